// GATModel_78194174591302
// MI455X (gfx1250) — compile-verified
//
#include <hip/hip_runtime.h>
#include <hip/hip_bf16.h>
#include <math.h>

#define N_NODES   50000
#define N_EDGES   1600000
#define F_IN      128
#define HID       32
#define HEADS     4
#define NEG_SLOPE 0.2f

typedef __attribute__((ext_vector_type(2))) float v2f;
typedef __attribute__((ext_vector_type(8))) float v8f;

// ---------------------------------------------------------------------------
// WMMA GEMM:  H[M,N] = A[M,K] * W[K,N]   (row-major, f32, full precision)
// One wave32 per 16x16 output tile, K stepped by 4 via V_WMMA_F32_16X16X4_F32.
// Requires M%16==0, N%16==0, K%4==0 (true for all calls: 50000/16=3125).
// Fragment layouts per CDNA5 ISA 7.12.2:
//   A 16x4 : lane L -> row m=L&15 ; VGPR0/1 hold K = k0+2*(L>>4)+{0,1}
//   B 4x16 : lane L -> col n=L&15 ; VGPR0/1 hold K = k0+2*(L>>4)+{0,1}
//   C/D    : lane L -> col n=L&15 ; VGPR v holds row m = v + 8*(L>>4)
// ---------------------------------------------------------------------------
__global__ __launch_bounds__(32) void gemm_f32_wmma(
    const float* __restrict__ A, const float* __restrict__ W,
    float* __restrict__ H, int M, int K, int N)
{
    const int lane = threadIdx.x & 31;
    const int half = lane >> 4;   // 0 = lanes 0..15, 1 = lanes 16..31
    const int l15  = lane & 15;
    const int m0   = blockIdx.x * 16;
    const int n0   = blockIdx.y * 16;

    const float* __restrict__ arow = A + (size_t)(m0 + l15) * K;

    v8f acc = {0.f, 0.f, 0.f, 0.f, 0.f, 0.f, 0.f, 0.f};
    for (int k0 = 0; k0 < K; k0 += 4) {
        const int ka = k0 + 2 * half;
        v2f a, b;
        a.x = arow[ka];
        a.y = arow[ka + 1];
        b.x = W[(size_t)ka       * N + n0 + l15];
        b.y = W[(size_t)(ka + 1) * N + n0 + l15];
        // 8 args: (neg_a, A, neg_b, B, c_mod, C, reuse_a, reuse_b)
        acc = __builtin_amdgcn_wmma_f32_16x16x4_f32(
            false, a, false, b, (short)0, acc, false, false);
    }
#pragma unroll
    for (int v = 0; v < 8; ++v)
        H[(size_t)(m0 + v + 8 * half) * N + n0 + l15] = acc[v];
}

// ---------------------------------------------------------------------------
// Per-node attention scalars: alpha_s/alpha_d = <h[n,h,:], a_{s,d}[h,:]>
// Also initializes segment-max to -inf and denom to 0 for this layer.
// ---------------------------------------------------------------------------
__global__ void alpha_init_kernel(
    const float* __restrict__ h,
    const float* __restrict__ avs, const float* __restrict__ avd,
    float* __restrict__ as, float* __restrict__ ad,
    float* __restrict__ mmax, float* __restrict__ denom,
    int n, int heads, int ch)
{
    int i = blockIdx.x * blockDim.x + threadIdx.x;   // index over n*heads
    if (i >= n * heads) return;
    int hh = i % heads;
    const float* __restrict__ hrow = h + (size_t)(i / heads) * heads * ch + hh * ch;
    const float* __restrict__ vs = avs + hh * ch;
    const float* __restrict__ vd = avd + hh * ch;
    float ss = 0.f, sd = 0.f;
    for (int c = 0; c < ch; ++c) {
        float v = hrow[c];
        ss = fmaf(v, vs[c], ss);
        sd = fmaf(v, vd[c], sd);
    }
    as[i] = ss;
    ad[i] = sd;
    mmax[i]  = -INFINITY;
    denom[i] = 0.f;
}

__global__ void zero_kernel(float* __restrict__ p, size_t cnt)
{
    size_t i = (size_t)blockIdx.x * blockDim.x + threadIdx.x;
    if (i < cnt) p[i] = 0.f;
}

// float atomic max via signed/unsigned integer ordering trick
__device__ __forceinline__ void atomicMaxF32(float* addr, float val)
{
    if (val >= 0.f) atomicMax((int*)addr, __float_as_int(val));
    else            atomicMin((unsigned int*)addr, __float_as_uint(val));
}

__device__ __forceinline__ void edge_nodes(const int* __restrict__ eidx,
                                           int e, int nE, int& s, int& d)
{
    if (e < nE) { s = eidx[e]; d = eidx[nE + e]; }
    else        { s = d = e - nE; }   // appended self-loops
}

__device__ __forceinline__ float leaky(float x)
{
    return x > 0.f ? x : NEG_SLOPE * x;
}

// pass 1: segment max of leaky_relu(alpha_s[src]+alpha_d[dst]) over dst
__global__ void edge_max_kernel(
    const int* __restrict__ eidx,
    const float* __restrict__ as, const float* __restrict__ ad,
    float* __restrict__ mmax, int nE, int n, int heads)
{
    int t = blockIdx.x * blockDim.x + threadIdx.x;
    int total = (nE + n) * heads;
    if (t >= total) return;
    int e = t / heads, hh = t - e * heads;
    int s, d; edge_nodes(eidx, e, nE, s, d);
    float x = leaky(as[s * heads + hh] + ad[d * heads + hh]);
    atomicMaxF32(&mmax[d * heads + hh], x);
}

// pass 2: denom[dst] += exp(e - m[dst])
__global__ void edge_sum_kernel(
    const int* __restrict__ eidx,
    const float* __restrict__ as, const float* __restrict__ ad,
    const float* __restrict__ mmax, float* __restrict__ denom,
    int nE, int n, int heads)
{
    int t = blockIdx.x * blockDim.x + threadIdx.x;
    int total = (nE + n) * heads;
    if (t >= total) return;
    int e = t / heads, hh = t - e * heads;
    int s, d; edge_nodes(eidx, e, nE, s, d);
    float x = leaky(as[s * heads + hh] + ad[d * heads + hh]);
    atomicAdd(&denom[d * heads + hh], __expf(x - mmax[d * heads + hh]));
}

// pass 3: out[dst] += h[src] * alpha   (alpha recomputed, 32-wide message)
__global__ void edge_msg_kernel(
    const int* __restrict__ eidx,
    const float* __restrict__ as, const float* __restrict__ ad,
    const float* __restrict__ mmax, const float* __restrict__ denom,
    const float* __restrict__ hin, float* __restrict__ out,
    int nE, int n, int heads, int ch)
{
    int t = blockIdx.x * blockDim.x + threadIdx.x;
    int total = (nE + n) * heads;
    if (t >= total) return;
    int e = t / heads, hh = t - e * heads;
    int s, d; edge_nodes(eidx, e, nE, s, d);
    float x = leaky(as[s * heads + hh] + ad[d * heads + hh]);
    float alpha = __expf(x - mmax[d * heads + hh]) / denom[d * heads + hh];

    const float4* __restrict__ src4 =
        (const float4*)(hin + (size_t)s * heads * ch + hh * ch);
    float* __restrict__ dstp = out + (size_t)d * heads * ch + hh * ch;
#pragma unroll
    for (int q = 0; q < HID / 4; ++q) {
        float4 v = src4[q];
        atomicAdd(dstp + 4 * q + 0, v.x * alpha);
        atomicAdd(dstp + 4 * q + 1, v.y * alpha);
        atomicAdd(dstp + 4 * q + 2, v.z * alpha);
        atomicAdd(dstp + 4 * q + 3, v.w * alpha);
    }
}

__global__ void bias_relu_kernel(float* __restrict__ h,
                                 const float* __restrict__ b, int n, int hc)
{
    int i = blockIdx.x * blockDim.x + threadIdx.x;
    if (i >= n * hc) return;
    float v = h[i] + b[i % hc];
    h[i] = v > 0.f ? v : 0.f;
}

// final classifier: [N,32] @ [32,2] + blin  (too skinny for WMMA; plain VALU)
__global__ void linear_out_kernel(
    const float* __restrict__ h, const float* __restrict__ Wl,
    const float* __restrict__ bl, float* __restrict__ out, int n)
{
    int i = blockIdx.x * blockDim.x + threadIdx.x;
    if (i >= n) return;
    const float* __restrict__ row = h + (size_t)i * HID;
    float o0 = bl[0], o1 = bl[1];
#pragma unroll
    for (int c = 0; c < HID; ++c) {
        float v = row[c];
        o0 = fmaf(v, Wl[2 * c + 0], o0);
        o1 = fmaf(v, Wl[2 * c + 1], o1);
    }
    out[2 * i + 0] = o0;
    out[2 * i + 1] = o1;
}

// ---------------------------------------------------------------------------
extern "C" void kernel_launch(void* const* d_in, const int* in_sizes, int n_in,
                              void* d_out, int out_size, void* d_ws, size_t ws_size,
                              hipStream_t stream)
{
    const float* x    = (const float*)d_in[0];
    const int*   eidx = (const int*)  d_in[1];
    const float* W1   = (const float*)d_in[2];
    const float* a1s  = (const float*)d_in[3];
    const float* a1d  = (const float*)d_in[4];
    const float* b1   = (const float*)d_in[5];
    const float* W2   = (const float*)d_in[6];
    const float* a2s  = (const float*)d_in[7];
    const float* a2d  = (const float*)d_in[8];
    const float* b2   = (const float*)d_in[9];
    const float* W3   = (const float*)d_in[10];
    const float* a3s  = (const float*)d_in[11];
    const float* a3d  = (const float*)d_in[12];
    const float* b3   = (const float*)d_in[13];
    const float* Wlin = (const float*)d_in[14];
    const float* blin = (const float*)d_in[15];

    // workspace layout (floats)
    float* bufA = (float*)d_ws;                       // N*128 node features
    float* bufB = bufA + (size_t)N_NODES * F_IN;      // N*128 node features
    float* as   = bufB + (size_t)N_NODES * F_IN;      // N*HEADS
    float* ad   = as + (size_t)N_NODES * HEADS;
    float* mm   = ad + (size_t)N_NODES * HEADS;
    float* dn   = mm + (size_t)N_NODES * HEADS;

    auto run_layer = [&](const float* in, int K, const float* Wm,
                         const float* avs, const float* avd, const float* bias,
                         int heads, int ch, float* hbuf, float* obuf) {
        const int N = heads * ch;
        dim3 ggrid(N_NODES / 16, N / 16);
        gemm_f32_wmma<<<ggrid, 32, 0, stream>>>(in, Wm, hbuf, N_NODES, K, N);

        int nh = N_NODES * heads;
        alpha_init_kernel<<<(nh + 255) / 256, 256, 0, stream>>>(
            hbuf, avs, avd, as, ad, mm, dn, N_NODES, heads, ch);

        size_t cnt = (size_t)N_NODES * N;
        zero_kernel<<<(unsigned)((cnt + 255) / 256), 256, 0, stream>>>(obuf, cnt);

        int totE = (N_EDGES + N_NODES) * heads;
        int eb = (totE + 255) / 256;
        edge_max_kernel<<<eb, 256, 0, stream>>>(eidx, as, ad, mm,
                                                N_EDGES, N_NODES, heads);
        edge_sum_kernel<<<eb, 256, 0, stream>>>(eidx, as, ad, mm, dn,
                                                N_EDGES, N_NODES, heads);
        edge_msg_kernel<<<eb, 256, 0, stream>>>(eidx, as, ad, mm, dn, hbuf, obuf,
                                                N_EDGES, N_NODES, heads, ch);
        bias_relu_kernel<<<(unsigned)((cnt + 255) / 256), 256, 0, stream>>>(
            obuf, bias, N_NODES, N);
    };

    // layer 1: x[50000,128] -> bufB[50000,128]
    run_layer(x,    F_IN, W1, a1s, a1d, b1, HEADS, HID, bufA, bufB);
    // layer 2: bufB -> bufB (gemm into bufA, aggregate back into bufB)
    run_layer(bufB, F_IN, W2, a2s, a2d, b2, HEADS, HID, bufA, bufB);
    // layer 3: single head, 32 channels
    run_layer(bufB, F_IN, W3, a3s, a3d, b3, 1,     HID, bufA, bufB);

    linear_out_kernel<<<(N_NODES + 255) / 256, 256, 0, stream>>>(
        bufB, Wlin, blin, (float*)d_out, N_NODES);
}